// LightGCN_14113262535120
// MI455X (gfx1250) — compile-verified
//
#include <hip/hip_runtime.h>
#include <hip/hip_bf16.h>

#define NUM_USERS 100000
#define NUM_ITEMS 200000
#define N_TOTAL   (NUM_USERS + NUM_ITEMS)
#define EMBED_DIM 64
#define NUM_LAYERS 3
#define NUM_EDGES 1200000
#define BATCH 4096

typedef __attribute__((ext_vector_type(2))) float v2f;
typedef __attribute__((ext_vector_type(8))) float v8f;

// ---------------------------------------------------------------------------
// Edge propagation: next[dst] += cur[src].  16 threads per edge, float4 each.
// cur rows are L2-resident (76.8MB < 192MB L2); atomics resolve in L2.
// ---------------------------------------------------------------------------
__global__ __launch_bounds__(256) void scatter_add_kernel(
    const float* __restrict__ cur, float* __restrict__ nxt,
    const int* __restrict__ src, const int* __restrict__ dst) {
  long long t = (long long)blockIdx.x * blockDim.x + threadIdx.x;
  int e = (int)(t >> 4);
  if (e >= NUM_EDGES) return;
  int part = ((int)t & 15) << 2;  // 0,4,...,60
  size_t s = (size_t)src[e];
  size_t d = (size_t)dst[e];
  const float4 v = *(const float4*)(cur + s * EMBED_DIM + part);
  float* p = nxt + d * EMBED_DIM + part;
  atomicAdd(p + 0, v.x);
  atomicAdd(p + 1, v.y);
  atomicAdd(p + 2, v.z);
  atomicAdd(p + 3, v.w);
}

// ---------------------------------------------------------------------------
// Gather batch rows of one layer output and accumulate into ubuf/vbuf.
// ubuf[b] += layer[uidx[b]] ; vbuf[b] += layer[NUM_USERS + iidx[b]]
// ---------------------------------------------------------------------------
__global__ __launch_bounds__(256) void gather_acc_kernel(
    const float* __restrict__ layer,
    const int* __restrict__ uidx, const int* __restrict__ iidx,
    float* __restrict__ ubuf, float* __restrict__ vbuf) {
  int t = blockIdx.x * blockDim.x + threadIdx.x;
  int b = t >> 4;
  if (b >= BATCH) return;
  int part = (t & 15) << 2;
  size_t u = (size_t)uidx[b];
  size_t it = (size_t)iidx[b] + NUM_USERS;
  const float4 uu = *(const float4*)(layer + u * EMBED_DIM + part);
  const float4 vv = *(const float4*)(layer + it * EMBED_DIM + part);
  float4* up = (float4*)(ubuf + (size_t)b * EMBED_DIM + part);
  float4* vp = (float4*)(vbuf + (size_t)b * EMBED_DIM + part);
  float4 a = *up;
  a.x += uu.x; a.y += uu.y; a.z += uu.z; a.w += uu.w;
  *up = a;
  float4 c = *vp;
  c.x += vv.x; c.y += vv.y; c.z += vv.z; c.w += vv.w;
  *vp = c;
}

// ---------------------------------------------------------------------------
// Batched dot products via WMMA: for each tile of 16 pairs, compute the
// 16x16 cross-product matrix U(16x64) x V^T(64x16) with 16 chained
// V_WMMA_F32_16X16X4_F32 ops and extract the diagonal.
// Fragment layout (ISA 7.12.2, 32-bit A 16x4): lane L holds
//   row M=L&15, K = (L>=16 ? 2 : 0) + {0,1}  (VGPR0=K even, VGPR1=K odd).
// B (4x16) is the mirror with N=L&15.  Diagonal of C/D:
//   m in 0..7  -> lane m,    VGPR m
//   m in 8..15 -> lane m+16, VGPR m-8
// ---------------------------------------------------------------------------
__global__ __launch_bounds__(128) void dot_wmma_kernel(
    const float* __restrict__ ubuf, const float* __restrict__ vbuf,
    float* __restrict__ out) {
  const int lane = threadIdx.x & 31;
  const int tile = blockIdx.x * (blockDim.x >> 5) + (threadIdx.x >> 5);
  const int rowBase = tile * 16;
  const int row = rowBase + (lane & 15);
  const int kOff = (lane >> 4) << 1;  // 0 for lanes 0-15, 2 for lanes 16-31

  const float* uRow = ubuf + (size_t)row * EMBED_DIM + kOff;
  const float* vRow = vbuf + (size_t)row * EMBED_DIM + kOff;

  v8f c = {};
#pragma unroll
  for (int k = 0; k < EMBED_DIM; k += 4) {
    v2f a, b;
    a.x = uRow[k];
    a.y = uRow[k + 1];
    b.x = vRow[k];
    b.y = vRow[k + 1];
    // 8 args: (neg_a, A, neg_b, B, c_mod, C, reuse_a, reuse_b)
    c = __builtin_amdgcn_wmma_f32_16x16x4_f32(false, a, false, b, (short)0, c,
                                              false, false);
  }

  // Extract diagonal element owned by this lane (if any).
  float cd[8];
#pragma unroll
  for (int i = 0; i < 8; ++i) cd[i] = c[i];
  int dsel = (lane < 16) ? lane : (lane - 24);  // valid for lane<8 or lane>=24
  float diag = cd[0];
#pragma unroll
  for (int i = 1; i < 8; ++i)
    if (dsel == i) diag = cd[i];

  const float scale = 1.0f / 16.0f;  // (acc/4)·(acc/4)
  if (lane < 8)
    out[rowBase + lane] = diag * scale;
  else if (lane >= 24)
    out[rowBase + (lane - 16)] = diag * scale;
}

extern "C" void kernel_launch(void* const* d_in, const int* in_sizes, int n_in,
                              void* d_out, int out_size, void* d_ws,
                              size_t ws_size, hipStream_t stream) {
  const float* emb = (const float*)d_in[0];
  const int* eidx = (const int*)d_in[1];
  const int* uidx = (const int*)d_in[2];
  const int* iidx = (const int*)d_in[3];
  float* out = (float*)d_out;

  const int* src = eidx;
  const int* dst = eidx + NUM_EDGES;

  char* ws = (char*)d_ws;
  const size_t nodeBytes = (size_t)N_TOTAL * EMBED_DIM * sizeof(float);
  const size_t batBytes = (size_t)BATCH * EMBED_DIM * sizeof(float);
  float* bufA = (float*)ws;
  float* bufB = (float*)(ws + nodeBytes);
  float* ubuf = (float*)(ws + 2 * nodeBytes);
  float* vbuf = (float*)(ws + 2 * nodeBytes + batBytes);

  // Layer-0 contribution: gather raw embeddings into fresh accumulators.
  hipMemsetAsync(ubuf, 0, batBytes, stream);
  hipMemsetAsync(vbuf, 0, batBytes, stream);
  const int gGrid = (BATCH * 16 + 255) / 256;
  gather_acc_kernel<<<gGrid, 256, 0, stream>>>(emb, uidx, iidx, ubuf, vbuf);

  // Three propagation layers (ping-pong bufA/bufB; layer 0 reads input).
  const long long sThreads = (long long)NUM_EDGES * 16;
  const int sGrid = (int)((sThreads + 255) / 256);
  const float* cur = emb;
  for (int l = 0; l < NUM_LAYERS; ++l) {
    float* nxt = (l & 1) ? bufB : bufA;
    hipMemsetAsync(nxt, 0, nodeBytes, stream);
    scatter_add_kernel<<<sGrid, 256, 0, stream>>>(cur, nxt, src, dst);
    gather_acc_kernel<<<gGrid, 256, 0, stream>>>(nxt, uidx, iidx, ubuf, vbuf);
    cur = nxt;
  }

  // 256 tiles of 16 pairs, one wave32 each, 4 waves per block.
  dot_wmma_kernel<<<BATCH / 16 / 4, 128, 0, stream>>>(ubuf, vbuf, out);
}